// Filter_13056700580349
// MI455X (gfx1250) — compile-verified
//
#include <hip/hip_runtime.h>
#include <hip/hip_bf16.h>
#include <stdint.h>

#define BATCH 32
#define NPRED 10647
#define NATTR 85
#define NCLS  80
#define SCORE_THRESH 0.3f
#define MAX_OUT 30
#define IOU_THRESH 0.5f
#define EPS_F 1e-9f

#define TILE 64                        // predictions per block, phase 1 (1 pred/thread)
#define TILE_BYTES (TILE * NATTR * 4)  // 21760 B of LDS staging
#define NMS_T 1024                     // threads per NMS block (32 wave32s)
#define NMS_W (NMS_T / 32)

// ---- CDNA5 async global->LDS copy (ASYNCcnt-tracked, bypasses VGPRs) ----
__device__ __forceinline__ void async_load_b128(uint32_t lds_off, uint32_t voff,
                                                const void* sbase) {
    asm volatile("global_load_async_to_lds_b128 %0, %1, %2"
                 :: "v"(lds_off), "v"(voff), "s"(sbase) : "memory");
}
__device__ __forceinline__ void async_load_b32(uint32_t lds_off, uint32_t voff,
                                               const void* sbase) {
    asm volatile("global_load_async_to_lds_b32 %0, %1, %2"
                 :: "v"(lds_off), "v"(voff), "s"(sbase) : "memory");
}
__device__ __forceinline__ void wait_async0() {
    asm volatile("s_wait_asynccnt 0" ::: "memory");
}

// =======================================================================
// Phase 1: decode. Stream (b, p0..p0+63, 0..84) tile into LDS via async
// copy, then one thread per prediction computes box / class / thresholded
// score. Record stride 85 floats -> per-lane LDS reads are conflict-free
// (85 mod 64 = 21, coprime with 64 banks).
// =======================================================================
__global__ __launch_bounds__(TILE) void yolo_decode_kernel(
    const float* __restrict__ preds,
    float4* __restrict__ boxes,       // [B*N] (y1,x1,y2,x2)
    float*  __restrict__ scores,      // [B*N]
    float*  __restrict__ classes)     // [B*N]
{
    __shared__ __align__(16) char smem[TILE_BYTES + 32];

    const int tid = threadIdx.x;
    const int b   = blockIdx.y;
    const int p0  = blockIdx.x * TILE;
    const int m   = (NPRED - p0 < TILE) ? (NPRED - p0) : TILE;

    // 16B-align the global base so the bulk copy can use B128 transfers.
    const uint64_t byte_base = ((uint64_t)b * NPRED + (uint64_t)p0) * (NATTR * 4ull);
    const uint64_t abase     = byte_base & ~15ull;
    const uint32_t shift     = (uint32_t)(byte_base - abase);     // 0,4,8,12
    const char*    gbase     = (const char*)preds + abase;

    const uint32_t nbytes = (uint32_t)m * (NATTR * 4u) + shift;
    const uint32_t nfull  = nbytes >> 4;          // full 16B chunks
    const uint32_t remdw  = (nbytes & 15u) >> 2;  // trailing dwords
    const uint32_t lds0   = (uint32_t)(size_t)(&smem[0]);

    for (uint32_t i = (uint32_t)tid; i < nfull; i += TILE)
        async_load_b128(lds0 + i * 16u, i * 16u, gbase);
    if ((uint32_t)tid < remdw)
        async_load_b32(lds0 + nfull * 16u + (uint32_t)tid * 4u,
                       nfull * 16u + (uint32_t)tid * 4u, gbase);
    wait_async0();          // each wave drains its own ASYNCcnt
    __syncthreads();        // then cross-wave LDS visibility

    if (tid < m) {
        const float* L = (const float*)(smem + shift + (size_t)tid * (NATTR * 4u));
        const float y1 = L[0], x1 = L[1], y2 = L[2], x2 = L[3];
        const float obj = L[4];

        float best = obj * L[5];
        int   bi   = 0;
        #pragma unroll 4
        for (int j = 1; j < NCLS; ++j) {
            const float s = obj * L[5 + j];
            if (s > best) { best = s; bi = j; }   // strict > keeps first max
        }
        const float sc = (best >= SCORE_THRESH) ? best : 0.0f;

        const int g = b * NPRED + p0 + tid;
        boxes[g]   = make_float4(y1, x1, y2, x2);
        scores[g]  = sc;
        classes[g] = (float)bi;
    }
}

// =======================================================================
// Phase 2: greedy NMS, one block per batch image. Both the mutable score
// vector (42.6 KB) and the full per-batch box array (170 KB) live in the
// WGP's 320 KB LDS (async-staged once) -- the 30 serial suppression scans
// then never leave the WGP. Argmax = wave32 shuffle reduction with
// first-index tie-break (matches jnp.argmax).
// =======================================================================
__global__ __launch_bounds__(NMS_T) void yolo_nms_kernel(
    const float4* __restrict__ boxes,
    const float*  __restrict__ scores,
    const float*  __restrict__ classes,
    float* __restrict__ out)          // [B, MAX_OUT, 6]
{
    __shared__ __align__(16) float4 s_bx[NPRED];   // 170352 B
    __shared__ float s_sc[NPRED];                  //  42588 B
    __shared__ float r_val[NMS_W];
    __shared__ int   r_idx[NMS_W];
    __shared__ float s_best[6];       // y1,x1,y2,x2,score,class
    __shared__ int   s_bi;

    const int tid = threadIdx.x;
    const int b   = blockIdx.x;
    const float*  sc_g = scores + (size_t)b * NPRED;
    const float4* bx_g = boxes  + (size_t)b * NPRED;

    {   // async-stage scores (b32) and boxes (b128) into LDS
        const uint32_t lds_sc = (uint32_t)(size_t)(&s_sc[0]);
        const uint32_t lds_bx = (uint32_t)(size_t)(&s_bx[0]);
        for (int i = tid; i < NPRED; i += NMS_T) {
            async_load_b32(lds_sc + (uint32_t)i * 4u, (uint32_t)i * 4u,
                           (const void*)sc_g);
            async_load_b128(lds_bx + (uint32_t)i * 16u, (uint32_t)i * 16u,
                            (const void*)bx_g);
        }
        wait_async0();
        __syncthreads();
    }

    const int lane = tid & 31;
    const int wid  = tid >> 5;

    for (int it = 0; it < MAX_OUT; ++it) {
        // ---- block-wide argmax over LDS scores ----
        float v  = -1.0f;
        int   ix = 0x7fffffff;
        for (int i = tid; i < NPRED; i += NMS_T) {
            const float s = s_sc[i];
            if (s > v) { v = s; ix = i; }
        }
        #pragma unroll
        for (int off = 16; off; off >>= 1) {
            const float ov = __shfl_down(v, off, 32);
            const int   oi = __shfl_down(ix, off, 32);
            if (ov > v || (ov == v && oi < ix)) { v = ov; ix = oi; }
        }
        if (lane == 0) { r_val[wid] = v; r_idx[wid] = ix; }
        __syncthreads();
        if (wid == 0) {
            float v2  = (lane < NMS_W) ? r_val[lane] : -1.0f;
            int   ix2 = (lane < NMS_W) ? r_idx[lane] : 0x7fffffff;
            #pragma unroll
            for (int off = 16; off; off >>= 1) {
                const float ov = __shfl_down(v2, off, 32);
                const int   oi = __shfl_down(ix2, off, 32);
                if (ov > v2 || (ov == v2 && oi < ix2)) { v2 = ov; ix2 = oi; }
            }
            if (lane == 0) {
                s_bi = ix2;
                const float4 bb = s_bx[ix2];
                s_best[0] = bb.x; s_best[1] = bb.y;
                s_best[2] = bb.z; s_best[3] = bb.w;
                s_best[4] = v2;
                s_best[5] = classes[(size_t)b * NPRED + ix2];
            }
        }
        __syncthreads();

        const float bestv = s_best[4];
        const bool  valid = bestv > 0.0f;

        if (tid == 0) {
            float* row = out + ((size_t)b * MAX_OUT + it) * 6;
            #pragma unroll
            for (int k = 0; k < 6; ++k) row[k] = valid ? s_best[k] : 0.0f;
        }

        if (valid) {
            const float by1 = s_best[0], bx1 = s_best[1];
            const float by2 = s_best[2], bx2 = s_best[3];
            const float area_a = (by2 - by1) * (bx2 - bx1);
            const int   bi = s_bi;
            for (int i = tid; i < NPRED; i += NMS_T) {
                const float4 q = s_bx[i];
                float ih = fminf(by2, q.z) - fmaxf(by1, q.x);
                float iw = fminf(bx2, q.w) - fmaxf(bx1, q.y);
                ih = fmaxf(ih, 0.0f);
                iw = fmaxf(iw, 0.0f);
                const float inter  = ih * iw;
                const float area_b = (q.z - q.x) * (q.w - q.y);
                const float iou    = inter / (area_a + area_b - inter + EPS_F);
                if (iou > IOU_THRESH || i == bi) s_sc[i] = 0.0f;
            }
        }
        __syncthreads();
    }
}

extern "C" void kernel_launch(void* const* d_in, const int* in_sizes, int n_in,
                              void* d_out, int out_size, void* d_ws, size_t ws_size,
                              hipStream_t stream) {
    (void)in_sizes; (void)n_in; (void)out_size; (void)ws_size;

    const float* preds = (const float*)d_in[0];
    float*       out   = (float*)d_out;

    // Workspace layout: boxes (float4[B*N]) | scores (float[B*N]) | classes (float[B*N])
    const size_t BN = (size_t)BATCH * NPRED;
    float4* boxesW   = (float4*)d_ws;
    float*  scoresW  = (float*)((char*)d_ws + BN * sizeof(float4));
    float*  classesW = scoresW + BN;

    dim3 g1((NPRED + TILE - 1) / TILE, BATCH);
    yolo_decode_kernel<<<g1, TILE, 0, stream>>>(preds, boxesW, scoresW, classesW);
    yolo_nms_kernel<<<BATCH, NMS_T, 0, stream>>>(boxesW, scoresW, classesW, out);
}